// MAG_SD_44796508897784
// MI455X (gfx1250) — compile-verified
//
#include <hip/hip_runtime.h>
#include <math.h>

// Problem constants (match reference setup_inputs).
#define BATCH 32
#define CH    3
#define IMG   480
#define AT    30
#define PAD   48   // int(0.1 * 480)

// Bilinear source coords for resizing 30 -> 480 (align_corners=False).
// src = (i+0.5)*(30/480) - 0.5, clipped to [0,29]; i1 = min(i0+1, 29).
__device__ __forceinline__ void coords480_from_30(int i, int& i0, int& i1, float& w) {
    float src = ((float)i + 0.5f) * 0.0625f - 0.5f;
    src = fminf(fmaxf(src, 0.0f), 29.0f);
    float f0 = floorf(src);
    float f1 = fminf(f0 + 1.0f, 29.0f);
    w  = src - f0;
    i0 = (int)f0;
    i1 = (int)f1;
}

// ---------------------------------------------------------------------------
// Kernel 1: per-sample threshold + bounding box.
// One block per batch sample. Stages the 30x30 attention tile into LDS via the
// CDNA5 async-DMA path (global_load_async_to_lds_b32, ASYNCcnt), then
// evaluates the 480x480 bilinear upsample entirely out of LDS.
// ---------------------------------------------------------------------------
__global__ __launch_bounds__(512)
void mag_bbox_kernel(const float* __restrict__ atten, int* __restrict__ bbox) {
    const int b   = blockIdx.x;
    const int tid = threadIdx.x;

    __shared__ float    sA[AT * AT];        // 30x30 attention tile
    __shared__ float    sRowI[IMG * AT];    // row-interp: [y][j], y=0..479, j=0..29
    __shared__ unsigned sRowAny[IMG];
    __shared__ unsigned sColAny[IMG];
    __shared__ float    sRed[512];
    __shared__ int      sMM[4];             // hmin, hmax, wmin, wmax

    const float* ap = atten + (size_t)b * (AT * AT);

    // --- Async DMA the attention tile into LDS (gfx1250 ASYNCcnt path) ---
    for (int i = tid; i < AT * AT; i += 512) {
        unsigned lds_off = (unsigned)(unsigned long long)(&sA[i]);  // low 32 bits = LDS offset
        const float* gp  = ap + i;
        asm volatile("global_load_async_to_lds_b32 %0, %1, off"
                     :: "v"(lds_off), "v"(gp) : "memory");
    }
    // Init flag arrays / min-max while the DMA is in flight.
    for (int i = tid; i < IMG; i += 512) { sRowAny[i] = 0u; sColAny[i] = 0u; }
    if (tid < 4) sMM[tid] = (tid & 1) ? -1 : IMG;   // {IMG, -1, IMG, -1}
    asm volatile("s_wait_asynccnt 0x0" ::: "memory");
    __syncthreads();

    // --- Block max-reduce over the 900 attention values ---
    float m = -1e30f;
    for (int i = tid; i < AT * AT; i += 512) m = fmaxf(m, sA[i]);
    sRed[tid] = m;
    __syncthreads();
    for (int s = 256; s > 0; s >>= 1) {
        if (tid < s) sRed[tid] = fmaxf(sRed[tid], sRed[tid + s]);
        __syncthreads();
    }
    const float theta = 0.5f * sRed[0];

    // --- Row interpolation first (matches reference op order) ---
    for (int idx = tid; idx < IMG * AT; idx += 512) {
        int y = idx / AT;
        int j = idx - y * AT;
        int r0, r1; float wr;
        coords480_from_30(y, r0, r1, wr);
        sRowI[idx] = sA[r0 * AT + j] * (1.0f - wr) + sA[r1 * AT + j] * wr;
    }
    __syncthreads();

    // --- Column interpolation + threshold -> row/col any flags ---
    for (int idx = tid; idx < IMG * IMG; idx += 512) {
        int y = idx / IMG;
        int x = idx - y * IMG;
        int c0, c1; float wc;
        coords480_from_30(x, c0, c1, wc);
        float up = sRowI[y * AT + c0] * (1.0f - wc) + sRowI[y * AT + c1] * wc;
        if (up >= theta) { sRowAny[y] = 1u; sColAny[x] = 1u; }
    }
    __syncthreads();

    // --- Reduce flags to min/max indices ---
    for (int i = tid; i < IMG; i += 512) {
        if (sRowAny[i]) { atomicMin(&sMM[0], i); atomicMax(&sMM[1], i); }
        if (sColAny[i]) { atomicMin(&sMM[2], i); atomicMax(&sMM[3], i); }
    }
    __syncthreads();
    if (tid == 0) {
        bbox[b * 4 + 0] = max(sMM[0] - PAD, 0);
        bbox[b * 4 + 1] = min(sMM[1] + PAD, IMG);
        bbox[b * 4 + 2] = max(sMM[2] - PAD, 0);
        bbox[b * 4 + 3] = min(sMM[3] + PAD, IMG);
    }
}

// ---------------------------------------------------------------------------
// Kernel 2: data-dependent crop + bilinear resize + blend (bandwidth-bound).
// One block per (b, c, y) output row; each thread produces 4 consecutive
// pixels with float4 (B128) center load + store. Gather taps hit L2 (entire
// 88.5 MB image tensor fits in the 192 MB L2).
// ---------------------------------------------------------------------------
__global__ __launch_bounds__(128)
void mag_crop_blend_kernel(const float* __restrict__ img,
                           const int*   __restrict__ bbox,
                           float*       __restrict__ out) {
    const int row = blockIdx.x;                 // B*CH*IMG rows
    const int y = row % IMG;
    const int c = (row / IMG) % CH;
    const int b = row / (IMG * CH);

    const int4 bb = *(const int4*)(bbox + b * 4);   // h0, h1, w0, w1
    const float cropH = (float)(bb.y - bb.x);
    const float cropW = (float)(bb.w - bb.z);
    const float sy = cropH / 480.0f;
    const float sx = cropW / 480.0f;

    // Row (vertical) source coords — uniform across the block.
    float srcy = ((float)y + 0.5f) * sy - 0.5f;
    srcy = fminf(fmaxf(srcy, 0.0f), cropH - 1.0f);
    float fy0 = floorf(srcy);
    float fy1 = fminf(fy0 + 1.0f, cropH - 1.0f);
    const float wy = srcy - fy0;
    const int r0 = min(max(bb.x + (int)fy0, 0), IMG - 1);
    const int r1 = min(max(bb.x + (int)fy1, 0), IMG - 1);

    const size_t planeOff = ((size_t)b * CH + c) * (size_t)(IMG * IMG);
    const float* plane = img + planeOff;
    const float* rowA  = plane + r0 * IMG;
    const float* rowB  = plane + r1 * IMG;
    const float* rowC  = plane + y  * IMG;
    float*       orow  = out + planeOff + (size_t)y * IMG;

    const int x0 = threadIdx.x * 4;
    if (x0 >= IMG) return;

    const float4 ctr = *(const float4*)(rowC + x0);
    const float cf[4] = {ctr.x, ctr.y, ctr.z, ctr.w};
    float res[4];

#pragma unroll
    for (int k = 0; k < 4; ++k) {
        int x = x0 + k;
        float srcx = ((float)x + 0.5f) * sx - 0.5f;
        srcx = fminf(fmaxf(srcx, 0.0f), cropW - 1.0f);
        float fx0 = floorf(srcx);
        float fx1 = fminf(fx0 + 1.0f, cropW - 1.0f);
        float wx = srcx - fx0;
        int c0 = min(max(bb.z + (int)fx0, 0), IMG - 1);
        int c1 = min(max(bb.z + (int)fx1, 0), IMG - 1);
        // Reference order: vertical interp first, then horizontal.
        float v0 = rowA[c0] * (1.0f - wy) + rowB[c0] * wy;
        float v1 = rowA[c1] * (1.0f - wy) + rowB[c1] * wy;
        float patch = v0 * (1.0f - wx) + v1 * wx;
        res[k] = cf[k] * 0.6f + patch * 0.4f;
    }
    *(float4*)(orow + x0) = make_float4(res[0], res[1], res[2], res[3]);
}

extern "C" void kernel_launch(void* const* d_in, const int* in_sizes, int n_in,
                              void* d_out, int out_size, void* d_ws, size_t ws_size,
                              hipStream_t stream) {
    (void)in_sizes; (void)n_in; (void)out_size; (void)ws_size;
    const float* images = (const float*)d_in[0];   // (32, 3, 480, 480) f32
    const float* atten  = (const float*)d_in[1];   // (32, 1, 30, 30)  f32
    float* out = (float*)d_out;                    // (32, 3, 480, 480) f32
    int*   bbox = (int*)d_ws;                      // 32 * int4

    mag_bbox_kernel<<<dim3(BATCH), dim3(512), 0, stream>>>(atten, bbox);
    mag_crop_blend_kernel<<<dim3(BATCH * CH * IMG), dim3(128), 0, stream>>>(images, bbox, out);
}